// GNN_Feature_Extractor_3805341024666
// MI455X (gfx1250) — compile-verified
//
#include <hip/hip_runtime.h>
#include <hip/hip_bf16.h>

// ---------------------------------------------------------------------------
// GAT feature extractor for MI455X (gfx1250), wave32, WMMA f32 16x16x4.
// ---------------------------------------------------------------------------

#define N_NODES   50000
#define N_EDGES   800000
#define E_TOTAL   850000     // + self loops
#define IN_F      128
#define HID       64
#define H1        4
#define C1        (H1*HID)   // 256
#define C2        HID        // 64
#define N_GRAPHS  32
#define NEG_SLOPE 0.2f

typedef __attribute__((ext_vector_type(2))) float v2f;
typedef __attribute__((ext_vector_type(8))) float v8f;

// ---- helpers ---------------------------------------------------------------

__device__ __forceinline__ void edge_sd(const int* __restrict__ ei, int e,
                                        int& s, int& d) {
  if (e < N_EDGES) { s = ei[e]; d = ei[N_EDGES + e]; }
  else             { s = d = e - N_EDGES; }          // self loop
}

__device__ __forceinline__ float leaky(float v) {
  return v > 0.0f ? v : NEG_SLOPE * v;
}

// monotone order-preserving float->uint encoding (for atomicMax on u32).
// encode(-inf)=0x007FFFFF > 0, so zero-init is below every real value.
__device__ __forceinline__ unsigned enc_f32(float f) {
  unsigned u = __float_as_uint(f);
  return (u & 0x80000000u) ? ~u : (u | 0x80000000u);
}
__device__ __forceinline__ float dec_f32(unsigned u) {
  unsigned b = (u & 0x80000000u) ? (u ^ 0x80000000u) : ~u;
  return __uint_as_float(b);
}

// ---- init kernels ----------------------------------------------------------

__global__ void zero_fill_k(unsigned* __restrict__ p, long long n) {
  long long i = (long long)blockIdx.x * blockDim.x + threadIdx.x;
  if (i < n) p[i] = 0u;
}

__global__ void bias_init_k(float* __restrict__ out,
                            const float* __restrict__ bias,
                            long long total, int k) {
  long long i = (long long)blockIdx.x * blockDim.x + threadIdx.x;
  if (i < total) out[i] = bias[i % k];
}

// ---- WMMA GEMM: H1 = X @ W1   [50000x128]x[128x256] ------------------------
// one wave32 per 16x16 tile. A: lane=M (lane&15), K pair split across lane
// halves (K = 2*(lane>>4)+j). B: lane=N, same K split. D: N=lane&15,
// M = 8*(lane>>4)+j for VGPR j.

__global__ __launch_bounds__(32) void gemm1_wmma(const float* __restrict__ X,
                                                 const float* __restrict__ W,
                                                 float* __restrict__ H) {
  const int tm = blockIdx.x;           // 0..3124
  const int tn = blockIdx.y;           // 0..15
  const int lane = threadIdx.x;        // 0..31
  const int mrow = (tm << 4) + (lane & 15);
  const int ncol = (tn << 4) + (lane & 15);
  const int khalf = (lane >> 4) << 1;  // 0 or 2
  const float* __restrict__ xrow = X + (size_t)mrow * IN_F;
  v8f acc = {};
#pragma unroll
  for (int k0 = 0; k0 < IN_F; k0 += 4) {
    v2f a, b;
    a.x = xrow[k0 + khalf + 0];
    a.y = xrow[k0 + khalf + 1];
    b.x = W[(size_t)(k0 + khalf + 0) * C1 + ncol];
    b.y = W[(size_t)(k0 + khalf + 1) * C1 + ncol];
    acc = __builtin_amdgcn_wmma_f32_16x16x4_f32(false, a, false, b,
                                                (short)0, acc, false, false);
  }
  const int mhalf = (lane >> 4) << 3;
#pragma unroll
  for (int j = 0; j < 8; ++j) {
    int m = (tm << 4) + mhalf + j;
    H[(size_t)m * C1 + ncol] = acc[j];
  }
}

// ---- WMMA GEMM 2: H2 = relu(OUT1) @ W2  [50000x256]x[256x64] ---------------

__global__ __launch_bounds__(32) void gemm2_wmma(const float* __restrict__ X,
                                                 const float* __restrict__ W,
                                                 float* __restrict__ H) {
  const int tm = blockIdx.x;           // 0..3124
  const int tn = blockIdx.y;           // 0..3
  const int lane = threadIdx.x;
  const int mrow = (tm << 4) + (lane & 15);
  const int ncol = (tn << 4) + (lane & 15);
  const int khalf = (lane >> 4) << 1;
  const float* __restrict__ xrow = X + (size_t)mrow * C1;
  v8f acc = {};
#pragma unroll
  for (int k0 = 0; k0 < C1; k0 += 4) {
    v2f a, b;
    a.x = fmaxf(xrow[k0 + khalf + 0], 0.0f);     // fused ReLU
    a.y = fmaxf(xrow[k0 + khalf + 1], 0.0f);
    b.x = W[(size_t)(k0 + khalf + 0) * C2 + ncol];
    b.y = W[(size_t)(k0 + khalf + 1) * C2 + ncol];
    acc = __builtin_amdgcn_wmma_f32_16x16x4_f32(false, a, false, b,
                                                (short)0, acc, false, false);
  }
  const int mhalf = (lane >> 4) << 3;
#pragma unroll
  for (int j = 0; j < 8; ++j) {
    int m = (tm << 4) + mhalf + j;
    H[(size_t)m * C2 + ncol] = acc[j];
  }
}

// ---- attention logits: per-node, per-head dot(h, a_src/a_dst) --------------

__global__ __launch_bounds__(256) void alpha1_k(const float* __restrict__ Hm,
                                                const float* __restrict__ a_src,
                                                const float* __restrict__ a_dst,
                                                float* __restrict__ as,
                                                float* __restrict__ ad) {
  __shared__ float ss[256], sd[256];
  const int n = blockIdx.x, t = threadIdx.x;      // a_* flat [4,64] matches t
  float hv = Hm[(size_t)n * C1 + t];
  ss[t] = hv * a_src[t];
  sd[t] = hv * a_dst[t];
  __syncthreads();
  for (int off = 32; off >= 1; off >>= 1) {
    if ((t & 63) < off) { ss[t] += ss[t + off]; sd[t] += sd[t + off]; }
    __syncthreads();
  }
  if ((t & 63) == 0) {
    as[n * H1 + (t >> 6)] = ss[t];
    ad[n * H1 + (t >> 6)] = sd[t];
  }
}

__global__ __launch_bounds__(64) void alpha2_k(const float* __restrict__ Hm,
                                               const float* __restrict__ a_src,
                                               const float* __restrict__ a_dst,
                                               float* __restrict__ as,
                                               float* __restrict__ ad) {
  __shared__ float ss[64], sd[64];
  const int n = blockIdx.x, t = threadIdx.x;
  float hv = Hm[(size_t)n * C2 + t];
  ss[t] = hv * a_src[t];
  sd[t] = hv * a_dst[t];
  __syncthreads();
  for (int off = 32; off >= 1; off >>= 1) {
    if (t < off) { ss[t] += ss[t + off]; sd[t] += sd[t + off]; }
    __syncthreads();
  }
  if (t == 0) { as[n] = ss[0]; ad[n] = sd[0]; }
}

// ---- segment softmax, layer 1 (H=4) ----------------------------------------

__global__ void edge_max1_k(const int* __restrict__ ei,
                            const float* __restrict__ as,
                            const float* __restrict__ ad,
                            unsigned* __restrict__ m1) {
  long long tid = (long long)blockIdx.x * blockDim.x + threadIdx.x;
  if (tid >= (long long)E_TOTAL * H1) return;
  int e = (int)(tid >> 2), h = (int)(tid & 3);
  int s, d; edge_sd(ei, e, s, d);
  float v = leaky(as[s * H1 + h] + ad[d * H1 + h]);
  atomicMax(&m1[d * H1 + h], enc_f32(v));
}

__global__ void edge_sum1_k(const int* __restrict__ ei,
                            const float* __restrict__ as,
                            const float* __restrict__ ad,
                            const unsigned* __restrict__ m1,
                            float* __restrict__ den1) {
  long long tid = (long long)blockIdx.x * blockDim.x + threadIdx.x;
  if (tid >= (long long)E_TOTAL * H1) return;
  int e = (int)(tid >> 2), h = (int)(tid & 3);
  int s, d; edge_sd(ei, e, s, d);
  float v = leaky(as[s * H1 + h] + ad[d * H1 + h]);
  float ex = __expf(v - dec_f32(m1[d * H1 + h]));
  atomicAdd(&den1[d * H1 + h], ex);
}

__global__ void edge_agg1_k(const int* __restrict__ ei,
                            const float* __restrict__ as,
                            const float* __restrict__ ad,
                            const unsigned* __restrict__ m1,
                            const float* __restrict__ den1,
                            const float* __restrict__ Hm,
                            float* __restrict__ out) {
  long long tid = (long long)blockIdx.x * blockDim.x + threadIdx.x;
  if (tid >= (long long)E_TOTAL * 64) return;    // 64 threads/edge, 4 floats ea
  int e = (int)(tid >> 6), q = (int)(tid & 63), h = q >> 4;
  int s, d; edge_sd(ei, e, s, d);
  float v = leaky(as[s * H1 + h] + ad[d * H1 + h]);
  float ex = __expf(v - dec_f32(m1[d * H1 + h]));
  float alpha = ex / (den1[d * H1 + h] + 1e-16f);
  const float4 hv = *(const float4*)(Hm + (size_t)s * C1 + q * 4);
  float* o = out + (size_t)d * C1 + q * 4;
  atomicAdd(o + 0, hv.x * alpha);
  atomicAdd(o + 1, hv.y * alpha);
  atomicAdd(o + 2, hv.z * alpha);
  atomicAdd(o + 3, hv.w * alpha);
}

// ---- segment softmax, layer 2 (H=1) ----------------------------------------

__global__ void edge_max2_k(const int* __restrict__ ei,
                            const float* __restrict__ as,
                            const float* __restrict__ ad,
                            unsigned* __restrict__ m2) {
  long long tid = (long long)blockIdx.x * blockDim.x + threadIdx.x;
  if (tid >= E_TOTAL) return;
  int e = (int)tid, s, d; edge_sd(ei, e, s, d);
  atomicMax(&m2[d], enc_f32(leaky(as[s] + ad[d])));
}

__global__ void edge_sum2_k(const int* __restrict__ ei,
                            const float* __restrict__ as,
                            const float* __restrict__ ad,
                            const unsigned* __restrict__ m2,
                            float* __restrict__ den2) {
  long long tid = (long long)blockIdx.x * blockDim.x + threadIdx.x;
  if (tid >= E_TOTAL) return;
  int e = (int)tid, s, d; edge_sd(ei, e, s, d);
  float ex = __expf(leaky(as[s] + ad[d]) - dec_f32(m2[d]));
  atomicAdd(&den2[d], ex);
}

__global__ void edge_agg2_k(const int* __restrict__ ei,
                            const float* __restrict__ as,
                            const float* __restrict__ ad,
                            const unsigned* __restrict__ m2,
                            const float* __restrict__ den2,
                            const float* __restrict__ Hm,
                            float* __restrict__ out) {
  long long tid = (long long)blockIdx.x * blockDim.x + threadIdx.x;
  if (tid >= (long long)E_TOTAL * 16) return;    // 16 threads/edge, 4 floats ea
  int e = (int)(tid >> 4), q = (int)(tid & 15);
  int s, d; edge_sd(ei, e, s, d);
  float ex = __expf(leaky(as[s] + ad[d]) - dec_f32(m2[d]));
  float alpha = ex / (den2[d] + 1e-16f);
  const float4 hv = *(const float4*)(Hm + (size_t)s * C2 + q * 4);
  float* o = out + (size_t)d * C2 + q * 4;
  atomicAdd(o + 0, hv.x * alpha);
  atomicAdd(o + 1, hv.y * alpha);
  atomicAdd(o + 2, hv.z * alpha);
  atomicAdd(o + 3, hv.w * alpha);
}

// ---- global mean pool ------------------------------------------------------

__global__ __launch_bounds__(64) void pool_k(const int* __restrict__ batch,
                                             const float* __restrict__ out2,
                                             float* __restrict__ psum,
                                             float* __restrict__ pcnt) {
  const int n = blockIdx.x, t = threadIdx.x;
  const int g = batch[n];
  atomicAdd(&psum[g * C2 + t], out2[(size_t)n * C2 + t]);
  if (t == 0) atomicAdd(&pcnt[g], 1.0f);
}

__global__ void finalize_k(const float* __restrict__ psum,
                           const float* __restrict__ pcnt,
                           float* __restrict__ out) {
  int i = blockIdx.x * blockDim.x + threadIdx.x;
  if (i < N_GRAPHS * C2) out[i] = psum[i] / fmaxf(pcnt[i / C2], 1.0f);
}

// ---------------------------------------------------------------------------

extern "C" void kernel_launch(void* const* d_in, const int* in_sizes, int n_in,
                              void* d_out, int out_size, void* d_ws, size_t ws_size,
                              hipStream_t stream) {
  const float* x      = (const float*)d_in[0];
  const int*   ei     = (const int*)  d_in[1];   // [2, E] flat
  const int*   batch  = (const int*)  d_in[2];
  const float* W1     = (const float*)d_in[3];
  const float* a_src1 = (const float*)d_in[4];
  const float* a_dst1 = (const float*)d_in[5];
  const float* b1     = (const float*)d_in[6];
  const float* W2     = (const float*)d_in[7];
  const float* a_src2 = (const float*)d_in[8];
  const float* a_dst2 = (const float*)d_in[9];
  const float* b2     = (const float*)d_in[10];
  float* out = (float*)d_out;

  // workspace layout (fp32 elements)
  float* ws = (float*)d_ws;
  float*    h1    = ws;                         // N*256
  float*    out1  = h1   + (size_t)N_NODES*C1;  // N*256
  float*    h2    = out1 + (size_t)N_NODES*C1;  // N*64
  float*    out2  = h2   + (size_t)N_NODES*C2;  // N*64
  float*    as1   = out2 + (size_t)N_NODES*C2;  // N*4
  float*    ad1   = as1  + (size_t)N_NODES*H1;  // N*4
  float*    den1  = ad1  + (size_t)N_NODES*H1;  // N*4  -- zero region start
  unsigned* m1    = (unsigned*)(den1 + (size_t)N_NODES*H1); // N*4
  float*    as2   = (float*)(m1 + (size_t)N_NODES*H1);      // N
  float*    ad2   = as2  + N_NODES;             // N
  float*    den2  = ad2  + N_NODES;             // N
  unsigned* m2    = (unsigned*)(den2 + N_NODES);// N
  float*    psum  = (float*)(m2 + N_NODES);     // 32*64
  float*    pcnt  = psum + N_GRAPHS*C2;         // 32   -- zero region end

  const long long nzero = (long long)N_NODES*H1*2 + (long long)N_NODES*4
                        + N_GRAPHS*C2 + N_GRAPHS;   // den1..pcnt contiguous
  zero_fill_k<<<(unsigned)((nzero + 255) / 256), 256, 0, stream>>>(
      (unsigned*)den1, nzero);
  bias_init_k<<<(unsigned)(((long long)N_NODES*C1 + 255) / 256), 256, 0, stream>>>(
      out1, b1, (long long)N_NODES*C1, C1);
  bias_init_k<<<(unsigned)(((long long)N_NODES*C2 + 255) / 256), 256, 0, stream>>>(
      out2, b2, (long long)N_NODES*C2, C2);

  // ---- layer 1 ----
  gemm1_wmma<<<dim3(N_NODES/16, C1/16), 32, 0, stream>>>(x, W1, h1);
  alpha1_k<<<N_NODES, 256, 0, stream>>>(h1, a_src1, a_dst1, as1, ad1);

  const long long eh1 = (long long)E_TOTAL * H1;
  edge_max1_k<<<(unsigned)((eh1 + 255) / 256), 256, 0, stream>>>(ei, as1, ad1, m1);
  edge_sum1_k<<<(unsigned)((eh1 + 255) / 256), 256, 0, stream>>>(ei, as1, ad1, m1, den1);
  const long long ew1 = (long long)E_TOTAL * 64;
  edge_agg1_k<<<(unsigned)((ew1 + 255) / 256), 256, 0, stream>>>(
      ei, as1, ad1, m1, den1, h1, out1);

  // ---- layer 2 (ReLU fused into GEMM2 A-load) ----
  gemm2_wmma<<<dim3(N_NODES/16, C2/16), 32, 0, stream>>>(out1, W2, h2);
  alpha2_k<<<N_NODES, 64, 0, stream>>>(h2, a_src2, a_dst2, as2, ad2);

  edge_max2_k<<<(E_TOTAL + 255) / 256, 256, 0, stream>>>(ei, as2, ad2, m2);
  edge_sum2_k<<<(E_TOTAL + 255) / 256, 256, 0, stream>>>(ei, as2, ad2, m2, den2);
  const long long ew2 = (long long)E_TOTAL * 16;
  edge_agg2_k<<<(unsigned)((ew2 + 255) / 256), 256, 0, stream>>>(
      ei, as2, ad2, m2, den2, h2, out2);

  // ---- global mean pool ----
  pool_k<<<N_NODES, 64, 0, stream>>>(batch, out2, psum, pcnt);
  finalize_k<<<(N_GRAPHS*C2 + 255) / 256, 256, 0, stream>>>(psum, pcnt, out);
}